// ConvNd_24893630447891
// MI455X (gfx1250) — compile-verified
//
#include <hip/hip_runtime.h>

// ---- f32 WMMA fragment types (wave32) ----
typedef __attribute__((ext_vector_type(2))) float v2f;  // A/B frag of 16x16x4 f32 (2 VGPRs)
typedef __attribute__((ext_vector_type(8))) float v8f;  // C/D frag 16x16 f32 (8 VGPRs)

// ---- problem constants (from reference) ----
#define BB_    2
#define CO_    4
#define ZIN_   64
#define WOUT_  30
#define XOUT_  30
#define YOUT_  62
#define ZOUT_  62
#define KPRIME 54          // CI * 3*3*3 (reduction after splitting last kernel dim into N)
#define NSTEPS 14          // ceil(54 / 4) WMMA k-steps
#define ZTILE  14          // complete outputs per 16-row tile
#define NZT    5           // ceil(62 / 14) z-tiles

// x element strides: (b, ci, w, x, y, z), z innermost
#define XS_Y   64
#define XS_X   4096
#define XS_W   131072
#define XS_CI  4194304
#define XS_B   8388608

// weight strides: (co, ci, a, b, c, d)
#define WS_D   1
#define WS_C   3
#define WS_B   9
#define WS_A   27
#define WS_CI  81
#define WS_CO  162

// ---- compile-time k' decode table: kp -> (x spatial offset, weight sub-index) ----
// kp in [0,54): kp = ((ci*3 + a)*3 + bk)*3 + c ; entries 54,55 clamp to kp=0 (B slots are 0)
#define KPC(k)  ((k) < KPRIME ? (k) : 0)
#define OFFE(k) ((KPC(k)/27)*XS_CI + ((KPC(k)%27)/9)*XS_W + ((KPC(k)%9)/3)*XS_X + (KPC(k)%3)*XS_Y)
#define WIDX(k) ((KPC(k)/27)*WS_CI + ((KPC(k)%27)/9)*WS_A + ((KPC(k)%9)/3)*WS_B + (KPC(k)%3)*WS_C)
#define KD(k)   { OFFE(k), WIDX(k) }

__constant__ int2 ktbl[56] = {
    KD(0),  KD(1),  KD(2),  KD(3),  KD(4),  KD(5),  KD(6),  KD(7),
    KD(8),  KD(9),  KD(10), KD(11), KD(12), KD(13), KD(14), KD(15),
    KD(16), KD(17), KD(18), KD(19), KD(20), KD(21), KD(22), KD(23),
    KD(24), KD(25), KD(26), KD(27), KD(28), KD(29), KD(30), KD(31),
    KD(32), KD(33), KD(34), KD(35), KD(36), KD(37), KD(38), KD(39),
    KD(40), KD(41), KD(42), KD(43), KD(44), KD(45), KD(46), KD(47),
    KD(48), KD(49), KD(50), KD(51), KD(52), KD(53), KD(54), KD(55)
};

__global__ __launch_bounds__(32)
void conv4d_wmma_f32(const float* __restrict__ x,
                     const float* __restrict__ w,
                     const float* __restrict__ bias,
                     float* __restrict__ out)
{
    __shared__ float lds[16 * 16];

    const int lane = threadIdx.x;       // 0..31 (wave32)
    const int n    = lane & 15;         // WMMA column index; also A row (t-row) index
    const int khi  = lane >> 4;         // which K-half this lane holds

    // ---- block coordinates: one block owns a full z-line (all 5 z-tiles) ----
    const int yo  = blockIdx.x;         // 0..61
    const int bwx = blockIdx.y;         // wo * XOUT + xo
    const int xo  = bwx % XOUT_;
    const int wo  = bwx / XOUT_;
    const int b   = blockIdx.z;

    // column n -> (co, d); columns 12..15 are dead (zero weights)
    const int  co   = n / 3;
    const int  d    = n % 3;
    const bool ncol = (n < 12);

    // uniform part of the x address lives in the scalar base pointer
    const float* xb = x + ((size_t)b * XS_B + wo * XS_W + xo * XS_X + yo * XS_Y);

    // ---- per-block setup via constant table (no div/mod chains) ----
    // Masks beyond the weight zeroing are unnecessary:
    //  * kp >= 54 pad elements have B == 0 for those k-slots, so A garbage * 0 = 0.
    //  * t >= 64 garbage rows land in D rows never read by the combine (z < 62 guard).
    const bool okA = ncol;                  // weight-live for steps 0..12
    const bool okB = ncol && (khi == 0);    // step 13: kp=52,53 live only in khi==0 lanes
    const int  wlb = co * WS_CO + d;        // per-lane weight base index

    v2f bf[NSTEPS];
    int va[NSTEPS][2];                      // per-lane load voffset (tile-invariant)
    #pragma unroll
    for (int s = 0; s < NSTEPS; ++s) {
        #pragma unroll
        for (int v = 0; v < 2; ++v) {
            const int2 kd = ktbl[4 * s + 2 * khi + v];
            va[s][v] = kd.x + n;
            const float wv = w[(unsigned)(wlb + kd.y)];
            const bool  ok = (s == NSTEPS - 1) ? okB : okA;
            bf[s][v] = ok ? wv : 0.0f;
        }
    }

    // ---- hoisted output bookkeeping: each lane writes <=2 of the 56 tile outputs ----
    const int  zi1 = lane % ZTILE;
    const int  cc1 = lane / ZTILE;                      // 0..2
    const int  i2  = lane + 32;
    const bool ok2 = (i2 < ZTILE * CO_);                // lane < 24
    const int  zi2 = i2 % ZTILE;
    const int  cc2 = ok2 ? (i2 / ZTILE) : 3;            // clamp so bias read stays in-bounds
    const float bias1 = bias[cc1];
    const float bias2 = bias[cc2];
    float* const op1 = out +
        (((((size_t)(b * CO_ + cc1) * WOUT_ + wo) * XOUT_ + xo) * YOUT_ + yo) * (size_t)ZOUT_ + zi1);
    float* const op2 = out +
        (((((size_t)(b * CO_ + cc2) * WOUT_ + wo) * XOUT_ + xo) * YOUT_ + yo) * (size_t)ZOUT_ + zi2);

    // last (partial) tile z-clamp: t = 56 + n -> 56 + min(n,7), applied as voffset delta
    const int dn = (n > 7) ? (n - 7) : 0;

    #pragma unroll
    for (int zt = 0; zt < NZT; ++zt) {
        const int  t0    = zt * ZTILE;                  // compile-time under full unroll
        const bool lastT = (zt == NZT - 1);

        // uniform scalar base bump: loads stay in SADDR + u32-voffset (scale_offset) form
        const float* xbt = xb + t0;

        // ---- load phase: 28 unconditional loads, zero per-tile vector address math ----
        float raw[NSTEPS][2];
        #pragma unroll
        for (int s = 0; s < NSTEPS; ++s) {
            #pragma unroll
            for (int v = 0; v < 2; ++v) {
                const int voff = lastT ? (va[s][v] - dn) : va[s][v];   // sub only in last tile
                raw[s][v] = xbt[(unsigned)voff];
            }
        }

        // ---- compute phase: 14 chained f32 WMMAs, no masking VALU ----
        v8f acc = {};
        #pragma unroll
        for (int s = 0; s < NSTEPS; ++s) {
            v2f af;
            af[0] = raw[s][0];
            af[1] = raw[s][1];
            acc = __builtin_amdgcn_wmma_f32_16x16x4_f32(
                      false, af, false, bf[s], (short)0, acc, false, false);
        }

        // ---- spill D (16x16) to LDS: lane holds rows m = v + 8*khi, col n ----
        #pragma unroll
        for (int v = 0; v < 8; ++v)
            lds[(v + 8 * khi) * 16 + n] = acc[v];
        __syncthreads();

        // ---- combine: out[co, z] = bias[co] + sum_d D[z - t0 + d, co*3 + d] ----
        {
            const int z = t0 + zi1;
            if (z < ZOUT_) {                            // compile-time true for zt < 4
                float r = bias1;
                #pragma unroll
                for (int dd = 0; dd < 3; ++dd)
                    r += lds[(zi1 + dd) * 16 + (cc1 * 3 + dd)];
                op1[t0] = r;                            // t0 folds into store imm offset
            }
        }
        if (ok2) {
            const int z = t0 + zi2;
            if (z < ZOUT_) {
                float r = bias2;
                #pragma unroll
                for (int dd = 0; dd < 3; ++dd)
                    r += lds[(zi2 + dd) * 16 + (cc2 * 3 + dd)];
                op2[t0] = r;
            }
        }
        __syncthreads();    // protect LDS before next tile overwrites it
    }
}

extern "C" void kernel_launch(void* const* d_in, const int* in_sizes, int n_in,
                              void* d_out, int out_size, void* d_ws, size_t ws_size,
                              hipStream_t stream) {
    const float* x    = (const float*)d_in[0];
    const float* wgt  = (const float*)d_in[1];
    const float* bias = (const float*)d_in[2];
    float* out        = (float*)d_out;

    dim3 grid(YOUT_,                // 62  : yo
              WOUT_ * XOUT_,        // 900 : wo * xo
              BB_);                 // 2   : batch
    dim3 block(32);                 // one wave32 per block
    conv4d_wmma_f32<<<grid, block, 0, stream>>>(x, wgt, bias, out);
}